// Decoder_66460323938638
// MI455X (gfx1250) — compile-verified
//
#include <hip/hip_runtime.h>
#include <math.h>

typedef float v2f __attribute__((ext_vector_type(2)));
typedef float v8f __attribute__((ext_vector_type(8)));
typedef unsigned int u32x4 __attribute__((ext_vector_type(4)));
typedef int i32x4 __attribute__((ext_vector_type(4)));
typedef int i32x8 __attribute__((ext_vector_type(8)));

constexpr int BB  = 32;     // batch
constexpr int TT  = 400;    // encoder frames
constexpr int EP  = 512;    // eprojs
constexpr int HID = 1024;   // dunits
constexpr int AD  = 512;    // att dim
constexpr int VOC = 5000;   // odim
constexpr int OL  = 100;    // decode steps
constexpr int G4  = 4 * HID;     // 4096
constexpr int DIN = HID + EP;    // 1536

constexpr int KCH  = 128;        // K chunk staged per TDM transfer
constexpr int LROW = KCH + 4;    // LDS row pitch in dwords (TDM pad: +4 dw / 128 dw)

__device__ __forceinline__ v8f wmma4(v8f acc, v2f a, v2f b) {
  // D = A(16x4 f32) * B(4x16 f32) + C, exact f32 accumulate
  return __builtin_amdgcn_wmma_f32_16x16x4_f32(false, a, false, b, (short)0, acc,
                                               false, false);
}

// ---------------------------------------------------------------------------
// TDM: DMA a 16-row x 128-col f32 tile (row stride = lda elements) from global
// into LDS at byte offset lds_off, padding each 128-dword row with 4 dwords so
// WMMA A-fragment reads from LDS hit distinct banks.
// D# layout per CDNA5 ISA (group0/group1; 2D tensor, so
// groups 2/3 are zero). 6-arg builtin form (clang-23 / therock HIP headers).
// ---------------------------------------------------------------------------
__device__ __forceinline__ void tdm_load_tile16(const float* src, int lda,
                                                unsigned lds_off) {
  unsigned long long ga = (unsigned long long)(uintptr_t)src;
  u32x4 g0;
  g0.x = 1u;                                    // count=1 (valid user descriptor)
  g0.y = lds_off;                               // lds_addr [63:32]
  g0.z = (unsigned)(ga & 0xFFFFFFFFu);          // global_addr [95:64]
  g0.w = (unsigned)((ga >> 32) & 0x1FFFFFFu)    // global_addr [120:96]
         | (2u << 30);                          // type=2 ("image")
  i32x8 g1;
  g1[0] = (2 << 16)                             // data_size = 4 bytes
        | (1 << 20)                             // pad_enable
        | (6 << 22)                             // pad_interval: every 128 dwords
        | (3 << 25);                            // pad_amount: 4 dwords
  g1[1] = (int)(((unsigned)lda & 0xFFFFu) << 16);          // tensor_dim0 lo16
  g1[2] = (int)(((unsigned)lda >> 16) | (16u << 16));      // dim0 hi | tensor_dim1=16
  g1[3] = (int)(128u << 16);                               // tile_dim0 = 128
  g1[4] = 16;                                              // tile_dim1 = 16 rows
  g1[5] = lda;                                             // tensor_dim0_stride lo32
  g1[6] = 0;
  g1[7] = 0;
  i32x4 z4 = {0, 0, 0, 0};
  i32x8 z8 = {0, 0, 0, 0, 0, 0, 0, 0};
  __builtin_amdgcn_tensor_load_to_lds(g0, g1, z4, z4, z8, 0);
}

// ---------------------------------------------------------------------------
// C[M,N] = A[M,K] @ B[K,N] (+ bias[n]); one wave per 16x16 tile (small GEMMs:
// pre_enc and z0@Wdec). M,N multiples of 16, K multiple of 4.
// ---------------------------------------------------------------------------
__global__ void gemm_nn_bias(const float* __restrict__ A, const float* __restrict__ B,
                             const float* __restrict__ bias, float* __restrict__ C,
                             int M, int N, int K) {
  int wave = (blockIdx.x * blockDim.x + threadIdx.x) >> 5;
  int lane = threadIdx.x & 31;
  int ntiles = N >> 4;
  int tiles = (M >> 4) * ntiles;
  if (wave >= tiles) return;
  int tm = wave / ntiles, tn = wave - tm * ntiles;
  int half = lane >> 4, idx = lane & 15;
  const float* Ap = A + (size_t)(tm * 16 + idx) * K;
  const float* Bp = B + (size_t)(tn * 16 + idx);
  v8f acc = {};
  for (int k = 0; k < K; k += 4) {
    int ka = k + half * 2;
    v2f a; a.x = Ap[ka];              a.y = Ap[ka + 1];
    v2f b; b.x = Bp[(size_t)ka * N];  b.y = Bp[(size_t)(ka + 1) * N];
    acc = wmma4(acc, a, b);
  }
  int col = tn * 16 + idx;
  float bv = bias ? bias[col] : 0.0f;
  float* Cp = C + (size_t)(tm * 16 + half * 8) * N + col;
#pragma unroll
  for (int r = 0; r < 8; ++r) Cp[(size_t)r * N] = acc[r] + bv;
}

// ---------------------------------------------------------------------------
// LSTM gates with TDM-staged A tiles:
//   G[32,4096] = X[32,Kx]@Wih^T + H[32,1024]@Whh^T + bih + bhh
// Block (256 thr, 8 waves) = one 16-row x 128-col strip of G. Wave 0 DMAs the
// shared 16xKCH A chunk into LDS, s_wait_tensorcnt, barrier, then all 8 waves
// run 32 WMMAs from LDS-A x global-B (per-lane B row is contiguous).
// ---------------------------------------------------------------------------
__global__ void gemm_gates(const float* __restrict__ X, int Kx,
                           const float* __restrict__ Hs,
                           const float* __restrict__ Wih,
                           const float* __restrict__ Whh,
                           const float* __restrict__ bih,
                           const float* __restrict__ bhh,
                           float* __restrict__ Gt) {
  __shared__ float Alds[16 * LROW];
  const int nstrips = G4 / 128;               // 32
  int strip = blockIdx.x % nstrips;
  int tm    = blockIdx.x / nstrips;           // 0..1
  int wv    = threadIdx.x >> 5;               // 0..7
  int lane  = threadIdx.x & 31;
  int half = lane >> 4, idx = lane & 15;
  int col = strip * 128 + wv * 16 + idx;
  const float* Wi = Wih + (size_t)col * Kx;   // B^T: lane's column = Wih row
  const float* Wh = Whh + (size_t)col * HID;
  v8f acc = {};

  for (int k0 = 0; k0 < Kx; k0 += KCH) {      // X @ Wih^T
    __syncthreads();
    if (wv == 0) {
      tdm_load_tile16(X + (size_t)(tm * 16) * Kx + k0, Kx, 0u);
      __builtin_amdgcn_s_wait_tensorcnt(0);
    }
    __syncthreads();
#pragma unroll 8
    for (int k = 0; k < KCH; k += 4) {
      int ka = k + half * 2;
      v2f a; a.x = Alds[idx * LROW + ka]; a.y = Alds[idx * LROW + ka + 1];
      v2f b; b.x = Wi[k0 + ka];           b.y = Wi[k0 + ka + 1];
      acc = wmma4(acc, a, b);
    }
  }
  for (int k0 = 0; k0 < HID; k0 += KCH) {     // H @ Whh^T
    __syncthreads();
    if (wv == 0) {
      tdm_load_tile16(Hs + (size_t)(tm * 16) * HID + k0, HID, 0u);
      __builtin_amdgcn_s_wait_tensorcnt(0);
    }
    __syncthreads();
#pragma unroll 8
    for (int k = 0; k < KCH; k += 4) {
      int ka = k + half * 2;
      v2f a; a.x = Alds[idx * LROW + ka]; a.y = Alds[idx * LROW + ka + 1];
      v2f b; b.x = Wh[k0 + ka];           b.y = Wh[k0 + ka + 1];
      acc = wmma4(acc, a, b);
    }
  }
  float bv = bih[col] + bhh[col];
  float* Gp = Gt + (size_t)(tm * 16 + half * 8) * G4 + col;
#pragma unroll
  for (int r = 0; r < 8; ++r) Gp[(size_t)r * G4] = acc[r] + bv;
}

// ---------------------------------------------------------------------------
// Final projection, TDM-staged A, N-edge handling (5000 % 16 != 0):
//   Y[3200,5000] = Z[3200,1536] @ Wout[1536,5000] + bout
// Block = 16-row x 128-col strip; edge waves clamp B loads, skip stores.
// ---------------------------------------------------------------------------
__global__ void gemm_out(const float* __restrict__ Z, const float* __restrict__ W,
                         const float* __restrict__ bias, float* __restrict__ Y) {
  __shared__ float Alds[16 * LROW];
  const int nstrips = (VOC + 127) / 128;      // 40
  int strip = blockIdx.x % nstrips;
  int tm    = blockIdx.x / nstrips;           // 0..199
  int wv    = threadIdx.x >> 5;
  int lane  = threadIdx.x & 31;
  int half = lane >> 4, idx = lane & 15;
  int col  = strip * 128 + wv * 16 + idx;
  int colc = col < VOC ? col : (VOC - 1);     // keep loads in-bounds
  const float* Wp = W + colc;
  v8f acc = {};

  for (int k0 = 0; k0 < DIN; k0 += KCH) {
    __syncthreads();
    if (wv == 0) {
      tdm_load_tile16(Z + (size_t)(tm * 16) * DIN + k0, DIN, 0u);
      __builtin_amdgcn_s_wait_tensorcnt(0);
    }
    __syncthreads();
#pragma unroll 8
    for (int k = 0; k < KCH; k += 4) {
      int ka = k + half * 2;
      v2f a; a.x = Alds[idx * LROW + ka];        a.y = Alds[idx * LROW + ka + 1];
      v2f b; b.x = Wp[(size_t)(k0 + ka) * VOC];  b.y = Wp[(size_t)(k0 + ka + 1) * VOC];
      acc = wmma4(acc, a, b);
    }
  }
  if (col < VOC) {
    float bv = bias[col];
    float* Yp = Y + (size_t)(tm * 16 + half * 8) * VOC + col;
#pragma unroll
    for (int r = 0; r < 8; ++r) Yp[(size_t)r * VOC] = acc[r] + bv;
  }
}

// ---------------------------------------------------------------------------
// Additive attention: e[t] = g . tanh(pre_enc[b,t,:] + d[b,:]); softmax over
// t < hlen[b]; att_c = sum_t w[t] * hpad[b,t,:]. One block per batch row.
// w[t]=0 beyond hlen, so unmasked hpad/pre_enc tails are harmless.
// ---------------------------------------------------------------------------
__global__ void attn_kernel(const float* __restrict__ pre_enc,
                            const float* __restrict__ dvec,
                            const float* __restrict__ gvec,
                            const float* __restrict__ hpad,
                            const int* __restrict__ hlen,
                            float* __restrict__ att_c,
                            float* __restrict__ z_all, int step) {
  int b = blockIdx.x;
  int tid = threadIdx.x;               // 256 threads
  __shared__ float w[TT];
  __shared__ float red[256];
  int L = hlen[b];

  for (int t = tid; t < TT; t += 256) {
    float s = 0.0f;
    if (t < L) {
      const float* pe = pre_enc + ((size_t)b * TT + t) * AD;
      const float* dv = dvec + (size_t)b * AD;
      for (int a = 0; a < AD; ++a) s += tanhf(pe[a] + dv[a]) * gvec[a];
    }
    w[t] = s;
  }
  __syncthreads();

  float m = -3.4e38f;
  for (int t = tid; t < TT; t += 256)
    if (t < L) m = fmaxf(m, w[t]);
  red[tid] = m;
  __syncthreads();
  for (int s2 = 128; s2 > 0; s2 >>= 1) {
    if (tid < s2) red[tid] = fmaxf(red[tid], red[tid + s2]);
    __syncthreads();
  }
  m = red[0];
  __syncthreads();

  float ssum = 0.0f;
  for (int t = tid; t < TT; t += 256) {
    float v = (t < L) ? expf(w[t] - m) : 0.0f;
    w[t] = v;
    ssum += v;
  }
  red[tid] = ssum;
  __syncthreads();
  for (int s2 = 128; s2 > 0; s2 >>= 1) {
    if (tid < s2) red[tid] += red[tid + s2];
    __syncthreads();
  }
  float inv = 1.0f / red[0];
  __syncthreads();
  for (int t = tid; t < TT; t += 256) w[t] *= inv;
  __syncthreads();

  for (int j = tid; j < EP; j += 256) {
    float acc = 0.0f;
    const float* hp = hpad + (size_t)b * TT * EP + j;
    for (int t = 0; t < TT; ++t) acc += w[t] * hp[(size_t)t * EP];
    att_c[b * EP + j] = acc;
    z_all[((size_t)b * OL + step) * DIN + HID + j] = acc;  // gen_feat='sc' tail
  }
}

// dec_in[b] = [ embed[ys[b,step]] ; att_c[b] ]
__global__ void build_decin(const float* __restrict__ embed, const int* __restrict__ ys,
                            const float* __restrict__ att_c, float* __restrict__ dec_in,
                            int step) {
  int i = blockIdx.x * blockDim.x + threadIdx.x;
  if (i >= BB * DIN) return;
  int b = i / DIN, j = i - b * DIN;
  float v;
  if (j < HID) v = embed[(size_t)ys[b * OL + step] * HID + j];
  else         v = att_c[b * EP + (j - HID)];
  dec_in[i] = v;
}

// gate order i,f,g,o (torch); updates h,c in place; optionally writes z_all head
__global__ void lstm_pointwise(const float* __restrict__ Gt, float* __restrict__ h,
                               float* __restrict__ c, float* __restrict__ z_all,
                               int step) {
  int i = blockIdx.x * blockDim.x + threadIdx.x;
  if (i >= BB * HID) return;
  int b = i / HID, j = i - b * HID;
  const float* g = Gt + (size_t)b * G4;
  float ig = 1.0f / (1.0f + expf(-g[j]));
  float fg = 1.0f / (1.0f + expf(-g[HID + j]));
  float gg = tanhf(g[2 * HID + j]);
  float og = 1.0f / (1.0f + expf(-g[3 * HID + j]));
  float cn = fg * c[i] + ig * gg;
  float hn = og * tanhf(cn);
  c[i] = cn;
  h[i] = hn;
  if (z_all) z_all[((size_t)b * OL + step) * DIN + j] = hn;
}

__global__ void zero_kernel(float* __restrict__ p, int n) {
  int i = blockIdx.x * blockDim.x + threadIdx.x;
  if (i < n) p[i] = 0.0f;
}

extern "C" void kernel_launch(void* const* d_in, const int* in_sizes, int n_in,
                              void* d_out, int out_size, void* d_ws, size_t ws_size,
                              hipStream_t stream) {
  (void)in_sizes; (void)n_in; (void)out_size; (void)ws_size;
  const float* hpad = (const float*)d_in[0];
  const int*   hlen = (const int*)d_in[1];
  const int*   ys   = (const int*)d_in[2];
  const float* embed= (const float*)d_in[3];
  const float* Wih0 = (const float*)d_in[4];
  const float* Whh0 = (const float*)d_in[5];
  const float* bih0 = (const float*)d_in[6];
  const float* bhh0 = (const float*)d_in[7];
  const float* Wih1 = (const float*)d_in[8];
  const float* Whh1 = (const float*)d_in[9];
  const float* bih1 = (const float*)d_in[10];
  const float* bhh1 = (const float*)d_in[11];
  const float* Wenc = (const float*)d_in[12];
  const float* benc = (const float*)d_in[13];
  const float* Wdec = (const float*)d_in[14];
  const float* gvec = (const float*)d_in[15];
  const float* Wout = (const float*)d_in[16];
  const float* bout = (const float*)d_in[17];

  float* ws = (float*)d_ws;
  float* pre_enc = ws;  ws += (size_t)BB * TT * AD;   // 6.55M
  float* dvec    = ws;  ws += BB * AD;
  float* attc    = ws;  ws += BB * EP;
  float* z0      = ws;  ws += BB * HID;               // z0,c0,z1,c1 contiguous
  float* c0      = ws;  ws += BB * HID;
  float* z1      = ws;  ws += BB * HID;
  float* c1      = ws;  ws += BB * HID;
  float* dec_in  = ws;  ws += BB * DIN;
  float* gates   = ws;  ws += BB * G4;
  float* z_all   = ws;  ws += (size_t)BB * OL * DIN;  // 4.9M
  float* Y = (float*)d_out;

  // zero LSTM states (z0..c1 are contiguous)
  {
    int n = 4 * BB * HID;
    zero_kernel<<<(n + 255) / 256, 256, 0, stream>>>(z0, n);
  }
  // pre_enc = hpad @ Wenc + benc  [B*T, A]
  {
    int tiles = (BB * TT / 16) * (AD / 16);
    gemm_nn_bias<<<(tiles * 32 + 255) / 256, 256, 0, stream>>>(
        hpad, Wenc, benc, pre_enc, BB * TT, AD, EP);
  }

  for (int step = 0; step < OL; ++step) {
    // d = z0 @ Wdec  [32,512]
    {
      int tiles = (BB / 16) * (AD / 16);
      gemm_nn_bias<<<(tiles * 32 + 255) / 256, 256, 0, stream>>>(
          z0, Wdec, nullptr, dvec, BB, AD, HID);
    }
    attn_kernel<<<BB, 256, 0, stream>>>(pre_enc, dvec, gvec, hpad, hlen,
                                        attc, z_all, step);
    build_decin<<<(BB * DIN + 255) / 256, 256, 0, stream>>>(embed, ys, attc,
                                                            dec_in, step);
    // layer 0: gates = dec_in@Wih0^T + z0@Whh0^T + b
    gemm_gates<<<(BB / 16) * (G4 / 128), 256, 0, stream>>>(
        dec_in, DIN, z0, Wih0, Whh0, bih0, bhh0, gates);
    lstm_pointwise<<<(BB * HID + 255) / 256, 256, 0, stream>>>(
        gates, z0, c0, (float*)nullptr, step);
    // layer 1: gates = z0@Wih1^T + z1@Whh1^T + b
    gemm_gates<<<(BB / 16) * (G4 / 128), 256, 0, stream>>>(
        z0, HID, z1, Wih1, Whh1, bih1, bhh1, gates);
    lstm_pointwise<<<(BB * HID + 255) / 256, 256, 0, stream>>>(
        gates, z1, c1, z_all, step);
  }

  // y_all = z_all @ Wout + bout
  gemm_out<<<(BB * OL / 16) * ((VOC + 127) / 128), 256, 0, stream>>>(
      z_all, Wout, bout, Y);
}